// CorrTorch_71098888618209
// MI455X (gfx1250) — compile-verified
//
#include <hip/hip_runtime.h>
#include <stdint.h>

typedef float v2f __attribute__((ext_vector_type(2)));
typedef float v8f __attribute__((ext_vector_type(8)));

#define C_DIM 64
#define D_DIM 64
#define H_DIM 128
#define W_DIM 128
#define CS  (D_DIM * H_DIM * W_DIM)  // channel stride, elements (1,048,576)
#define CSB ((uint32_t)CS * 4u)      // channel stride, bytes (4 MB, fits imm24)
#define HW  (H_DIM * W_DIM)
#define DHW (D_DIM * H_DIM * W_DIM)  // output channel stride
#define TILES_X 9                    // ceil(128 / 15): stride-15 banded tiling
#define WAVES_PER_BLOCK 4
#define TOTAL_WAVES (TILES_X * H_DIM * D_DIM)  // 73728

__global__ __launch_bounds__(WAVES_PER_BLOCK * 32)
void corr27_wmma_kernel(const float* __restrict__ in1,
                        const float* __restrict__ in2,
                        float* __restrict__ out)
{
    __shared__ float lds[WAVES_PER_BLOCK][9 * 256];

    const int l  = threadIdx.x & 31;
    const int wv = threadIdx.x >> 5;
    const int gw = blockIdx.x * WAVES_PER_BLOCK + wv;

    const int tx = gw % TILES_X;
    const int y  = (gw / TILES_X) % H_DIM;
    const int z  =  gw / (TILES_X * H_DIM);

    const int x0   = tx * 15;
    const int m    = l & 15;       // A row == B col index within tile (both = lane&15)
    const int half = l >> 4;       // K-half selector for f32 WMMA operand layout
    const int xg   = x0 + m;
    const bool okX = xg < W_DIM;
    const int  xc  = okX ? xg : (W_DIM - 1);   // clamped, always-safe address

    const char* p1 = (const char*)in1;
    const char* p2 = (const char*)in2;

    // 32-bit BYTE offsets from the uniform bases -> GVS global_load_b32 with the
    // 4 MB channel-pair delta folded into the signed-24-bit instruction offset.
    uint32_t offA = ((uint32_t)(2 * half) * CS + (uint32_t)z * HW
                   + (uint32_t)y * W_DIM + (uint32_t)xc) * 4u;

    uint32_t offB[9];
    bool selB[9];
    #pragma unroll
    for (int i = 0; i < 9; ++i) {
        int zz = z + (i / 3) - 1;
        int yy = y + (i % 3) - 1;
        bool v = (zz >= 0) & (zz < D_DIM) & (yy >= 0) & (yy < H_DIM);
        selB[i] = okX & v;
        if (!v) { zz = 0; yy = 0; }  // keep offset in-bounds; value is select-zeroed
        offB[i] = ((uint32_t)(2 * half) * CS + (uint32_t)zz * HW
                 + (uint32_t)yy * W_DIM + (uint32_t)xc) * 4u;
    }

    v8f acc[9];
    #pragma unroll
    for (int i = 0; i < 9; ++i) acc[i] = v8f{};

    // K loop: 4 channels per V_WMMA_F32_16X16X4_F32 step.
    // Phase 1: issue all 20 loads (one clause, deep LOADcnt pipeline).
    // Phase 2: 9 WMMAs drain the loads with descending s_wait_loadcnt.
    // No control flow around WMMA -> EXEC stays all-ones as the ISA requires.
    for (int s = 0; s < C_DIM / 4; ++s) {
        float a0 = *(const float*)(p1 + offA);
        float a1 = *(const float*)(p1 + offA + CSB);
        offA += 4u * CSB;
        float b0[9], b1[9];
        #pragma unroll
        for (int i = 0; i < 9; ++i) {
            b0[i] = *(const float*)(p2 + offB[i]);
            b1[i] = *(const float*)(p2 + offB[i] + CSB);
            offB[i] += 4u * CSB;
        }
        v2f a;
        a.x = okX ? a0 : 0.0f;
        a.y = okX ? a1 : 0.0f;
        #pragma unroll
        for (int i = 0; i < 9; ++i) {
            v2f b;
            b.x = selB[i] ? b0[i] : 0.0f;
            b.y = selB[i] ? b1[i] : 0.0f;
            // (neg_a, A, neg_b, B, c_mod, C, reuse_a, reuse_b)
            acc[i] = __builtin_amdgcn_wmma_f32_16x16x4_f32(
                false, a, false, b, (short)0, acc[i], false, false);
        }
    }

    // Diagonal harvest: all 9 G tiles -> LDS -> 27 band outputs per x, one barrier.
    // C/D layout: VGPR r, lane l holds (M = r + 8*(l>>4), N = l&15).
    #pragma unroll
    for (int i = 0; i < 9; ++i)
        #pragma unroll
        for (int r = 0; r < 8; ++r)
            lds[wv][i * 256 + (r + 8 * half) * 16 + m] = acc[i][r];
    __syncthreads();

    const float scale = 1.0f / (float)C_DIM;
    if (l < 16 && (x0 + l) < W_DIM) {
        const int x = x0 + l;
        #pragma unroll 1
        for (int i = 0; i < 9; ++i) {
            uint32_t base = (uint32_t)(i * 3) * DHW + (uint32_t)z * HW
                          + (uint32_t)y * W_DIM + (uint32_t)x;
            const float* g = &lds[wv][i * 256];
            // dx = -1 (channel 3i+0): rows 1..15; x==0 is the padded border (tile 0)
            if (l >= 1)
                out[base] = g[l * 16 + (l - 1)] * scale;
            else if (tx == 0)
                out[base] = 0.0f;
            // dx = 0 (channel 3i+1): rows 0..14 (disjoint stride-15 partition)
            if (l <= 14)
                out[base + (uint32_t)DHW] = g[l * 16 + l] * scale;
            // dx = +1 (channel 3i+2): rows 0..14; x'==128 column was zero-filled
            if (l <= 14)
                out[base + 2u * (uint32_t)DHW] = g[l * 16 + (l + 1)] * scale;
        }
    }
}

extern "C" void kernel_launch(void* const* d_in, const int* in_sizes, int n_in,
                              void* d_out, int out_size, void* d_ws, size_t ws_size,
                              hipStream_t stream) {
    const float* in1 = (const float*)d_in[0];
    const float* in2 = (const float*)d_in[1];
    float* out = (float*)d_out;
    (void)in_sizes; (void)n_in; (void)out_size; (void)d_ws; (void)ws_size;

    dim3 block(WAVES_PER_BLOCK * 32);
    dim3 grid(TOTAL_WAVES / WAVES_PER_BLOCK);  // 18432 blocks, exact
    corr27_wmma_kernel<<<grid, block, 0, stream>>>(in1, in2, out);
}